// GNNCell_33517924778635
// MI455X (gfx1250) — compile-verified
//
#include <hip/hip_runtime.h>
#include <math.h>

// Problem constants (from reference setup_inputs): B=16, S=1024, D=300, 3 layers.
#define BB 16
#define SS 1024
#define DD 300
#define NEG_SLOPE 0.01f

typedef __attribute__((ext_vector_type(2))) float v2f;
typedef __attribute__((ext_vector_type(8))) float v8f;

// CDNA5 fp32 WMMA: D(16x16,f32) = A(16x4,f32) * B(4x16,f32) + C
// 8-arg form: (neg_a, A, neg_b, B, c_mod, C, reuse_a, reuse_b)
__device__ __forceinline__ v8f wmma_f32(v2f a, v2f b, v8f c) {
  return __builtin_amdgcn_wmma_f32_16x16x4_f32(false, a, false, b, (short)0, c,
                                               false, false);
}

// -------------------------------------------------------------------------
// Row squared-norms: one wave32 per row of (B*S, D).
// -------------------------------------------------------------------------
__global__ void __launch_bounds__(256)
row_norm_kernel(const float* __restrict__ x, float* __restrict__ out) {
  int gid  = blockIdx.x * blockDim.x + threadIdx.x;
  int row  = gid >> 5;
  int lane = gid & 31;
  const float* r = x + (size_t)row * DD;
  float s = 0.0f;
  for (int k = lane; k < DD; k += 32) { float v = r[k]; s += v * v; }
#pragma unroll
  for (int off = 16; off > 0; off >>= 1) s += __shfl_xor(s, off, 32);
  if (lane == 0) out[row] = s;
}

// -------------------------------------------------------------------------
// Fused GNN layer for one 16-row i-block per workgroup (8 waves).
// Never materializes s: per 128-wide j-chunk, phase 1 builds s and s^T
// 16x16 tiles with fp32 WMMA + sqrt epilogue into LDS; phase 2 consumes
// them as A-operands for the four update GEMMs. Epilogue: W*upd + Bias*x,
// leaky_relu.
// -------------------------------------------------------------------------
__global__ void __launch_bounds__(256)
gnn_layer_kernel(const float* __restrict__ h_in, const float* __restrict__ d_in,
                 const float* __restrict__ h2,   const float* __restrict__ d2,
                 const float* __restrict__ W1,   const float* __restrict__ W2,
                 const float* __restrict__ B1,   const float* __restrict__ B2,
                 float* __restrict__ h_out,      float* __restrict__ d_out) {
  // Row stride 20 words: conflict-free for C-layout stores and A-operand
  // float2 reads (row*20 mod 64 spans of 4 words are pairwise disjoint).
  __shared__ float lds_s[8][16][20];
  __shared__ float lds_t[8][16][20];

  const int w    = threadIdx.x >> 5;   // wave id 0..7
  const int lane = threadIdx.x & 31;
  const int mn   = lane & 15;          // M (A) / N (B,C) index
  const int hi   = lane >> 4;          // half-wave select
  const int hi2  = hi * 2;             // K sub-offset for A/B operands

  const int b  = blockIdx.y;
  const int i0 = blockIdx.x * 16;

  const float* hb  = h_in + (size_t)b * SS * DD;
  const float* db  = d_in + (size_t)b * SS * DD;
  const float* h2b = h2 + b * SS;
  const float* d2b = d2 + b * SS;

  // Norms for this block's 16 i-rows, mapped to the C layout (M = r + hi*8).
  float h2i[8], d2i[8];
#pragma unroll
  for (int r = 0; r < 8; ++r) {
    h2i[r] = h2b[i0 + r + hi * 8];
    d2i[r] = d2b[i0 + r + hi * 8];
  }

  // c-tile ownership: 19 tiles of 16 cols cover D=300; wave w owns tiles
  // w, w+8, w+16 (invalid ones masked out wave-uniformly).
  int  cbase[3]; bool tval[3]; int ccl[3];
#pragma unroll
  for (int t = 0; t < 3; ++t) {
    cbase[t] = (w + t * 8) * 16;
    tval[t]  = cbase[t] < DD;
    int c    = cbase[t] + mn;
    ccl[t]   = c < DD ? c : (DD - 1);  // clamped load col (garbage cols unstored)
  }

  v8f accH[3] = {{0}}, accD[3] = {{0}};

  // Phase-1 A operands: h rows i0.. (for s) and d rows i0.. (for s^T).
  const float* aS = hb + (size_t)(i0 + mn) * DD + hi2;
  const float* aT = db + (size_t)(i0 + mn) * DD + hi2;

  for (int chunk = 0; chunk < 8; ++chunk) {
    const int jbase = chunk * 128;
    const int jw    = jbase + w * 16;  // this wave's 16-wide j sub-chunk

    // ---- phase 1: cross GEMMs over K=D=300 (75 fp32 WMMA steps each) ----
    const float* bS = db + (size_t)(jw + mn) * DD + hi2;  // d rows j (B op)
    const float* bT = hb + (size_t)(jw + mn) * DD + hi2;  // h rows j (B op)
    v8f accS = {0}, accT = {0};
    for (int k = 0; k < DD; k += 4) {
      v2f a0 = *(const v2f*)(aS + k);
      v2f b0 = *(const v2f*)(bS + k);
      v2f a1 = *(const v2f*)(aT + k);
      v2f b1 = *(const v2f*)(bT + k);
      accS = wmma_f32(a0, b0, accS);   // dot(h_i, d_j)
      accT = wmma_f32(a1, b1, accT);   // dot(d_i, h_j)
    }
    const float h2j = h2b[jw + mn];
    const float d2j = d2b[jw + mn];

    __syncthreads();  // previous phase 2 is done reading LDS
#pragma unroll
    for (int r = 0; r < 8; ++r) {
      int m = r + hi * 8;
      float vs = h2i[r] + d2j - 2.0f * accS[r];          // s[i0+m, jw+mn]
      lds_s[w][m][mn] = vs > 0.0f ? sqrtf(vs) : 0.0f;
      float vt = d2i[r] + h2j - 2.0f * accT[r];          // s[jw+mn, i0+m]
      lds_t[w][m][mn] = vt > 0.0f ? sqrtf(vt) : 0.0f;
    }
    __syncthreads();

    // ---- phase 2: accH += s*h + sT*d ; accD += sT*h + s*d over this chunk
    for (int jj = 0; jj < 128; jj += 4) {
      const int wsel = jj >> 4;
      const int kb   = (jj & 15) + hi2;
      v2f as = *(const v2f*)&lds_s[wsel][mn][kb];
      v2f at = *(const v2f*)&lds_t[wsel][mn][kb];
      const int jk = jbase + jj + hi2;
      const float* hr = hb + (size_t)jk * DD;
      const float* dr = db + (size_t)jk * DD;
#pragma unroll
      for (int t = 0; t < 3; ++t) {
        if (!tval[t]) continue;  // wave-uniform
        v2f bh = { hr[ccl[t]], hr[DD + ccl[t]] };
        v2f bd = { dr[ccl[t]], dr[DD + ccl[t]] };
        accH[t] = wmma_f32(as, bh, accH[t]);
        accH[t] = wmma_f32(at, bd, accH[t]);
        accD[t] = wmma_f32(at, bh, accD[t]);
        accD[t] = wmma_f32(as, bd, accD[t]);
      }
    }
  }

  // ---- epilogue: W*update + Bias*x, leaky_relu ----
#pragma unroll
  for (int t = 0; t < 3; ++t) {
    if (!tval[t]) continue;
#pragma unroll
    for (int r = 0; r < 8; ++r) {
      int row = i0 + r + hi * 8;
      int c   = cbase[t] + mn;
      if (c < DD) {
        size_t pix = (size_t)row * DD + c;               // (S,D) params
        size_t gix = ((size_t)b * SS + row) * DD + c;    // (B,S,D) tensors
        float xh = W1[pix] * accH[t][r] + B1[pix] * h_in[gix];
        h_out[gix] = xh > 0.0f ? xh : NEG_SLOPE * xh;
        float xd = W2[pix] * accD[t][r] + B2[pix] * d_in[gix];
        d_out[gix] = xd > 0.0f ? xd : NEG_SLOPE * xd;
      }
    }
  }
}

// -------------------------------------------------------------------------
// Final pairwise distance: one 16x16 tile of (B,S,S) per wave.
// -------------------------------------------------------------------------
__global__ void __launch_bounds__(256)
dist_kernel(const float* __restrict__ h, const float* __restrict__ d,
            const float* __restrict__ h2, const float* __restrict__ d2,
            float* __restrict__ out) {
  const int w    = threadIdx.x >> 5;
  const int lane = threadIdx.x & 31;
  const int mn   = lane & 15;
  const int hi   = lane >> 4;
  const int hi2  = hi * 2;
  const int b  = blockIdx.z;
  const int i0 = blockIdx.y * 16;
  const int j0 = (blockIdx.x * 8 + w) * 16;

  const float* hb = h + (size_t)b * SS * DD;
  const float* db = d + (size_t)b * SS * DD;
  const float* arow = hb + (size_t)(i0 + mn) * DD + hi2;
  const float* brow = db + (size_t)(j0 + mn) * DD + hi2;

  v8f acc = {0};
  for (int k = 0; k < DD; k += 4) {
    v2f a  = *(const v2f*)(arow + k);
    v2f bv = *(const v2f*)(brow + k);
    acc = wmma_f32(a, bv, acc);
  }

  const float* h2b = h2 + b * SS;
  const float* d2b = d2 + b * SS;
  const float d2j = d2b[j0 + mn];
  float* ob = out + ((size_t)b * SS + i0) * SS + j0;
#pragma unroll
  for (int r = 0; r < 8; ++r) {
    int m = r + hi * 8;
    float v = h2b[i0 + m] + d2j - 2.0f * acc[r];
    ob[(size_t)m * SS + mn] = v > 0.0f ? sqrtf(v) : 0.0f;
  }
}

// -------------------------------------------------------------------------
// Host launch: 3 fused layers (ping-pong in workspace), then final distance.
// Workspace use: 4 * B*S*D floats (78.6 MB) + 2 * B*S floats (norms).
// -------------------------------------------------------------------------
extern "C" void kernel_launch(void* const* d_in, const int* in_sizes, int n_in,
                              void* d_out, int out_size, void* d_ws, size_t ws_size,
                              hipStream_t stream) {
  (void)in_sizes; (void)n_in; (void)out_size; (void)ws_size;
  const float* h0 = (const float*)d_in[0];  // head_embed
  const float* d0 = (const float*)d_in[1];  // dep_embed
  const float* W1 = (const float*)d_in[2];
  const float* W2 = (const float*)d_in[3];
  const float* B1 = (const float*)d_in[4];
  const float* B2 = (const float*)d_in[5];
  float* out = (float*)d_out;

  const size_t embN = (size_t)BB * SS * DD;
  float* ws    = (float*)d_ws;
  float* bufAh = ws;
  float* bufAd = bufAh + embN;
  float* bufBh = bufAd + embN;
  float* bufBd = bufBh + embN;
  float* h2    = bufBd + embN;
  float* d2    = h2 + (size_t)BB * SS;

  const dim3 blk(256);
  const dim3 gN((BB * SS * 32) / 256);       // one wave per row
  const dim3 gU(SS / 16, BB);                // i-tiles x batch
  const dim3 gD(SS / 16 / 8, SS / 16, BB);   // j-tile groups x i-tiles x batch

  const float* hc = h0;
  const float* dc = d0;
  float* outs[3][2] = { {bufAh, bufAd}, {bufBh, bufBd}, {bufAh, bufAd} };
  for (int layer = 0; layer < 3; ++layer) {
    row_norm_kernel<<<gN, blk, 0, stream>>>(hc, h2);
    row_norm_kernel<<<gN, blk, 0, stream>>>(dc, d2);
    gnn_layer_kernel<<<gU, blk, 0, stream>>>(hc, dc, h2, d2, W1, W2, B1, B2,
                                             outs[layer][0], outs[layer][1]);
    hc = outs[layer][0];
    dc = outs[layer][1];
  }
  row_norm_kernel<<<gN, blk, 0, stream>>>(hc, h2);
  row_norm_kernel<<<gN, blk, 0, stream>>>(dc, d2);
  dist_kernel<<<gD, blk, 0, stream>>>(hc, dc, h2, d2, out);
}